// GraphAttention_73452530696570
// MI455X (gfx1250) — compile-verified
//
#include <hip/hip_runtime.h>

typedef __attribute__((ext_vector_type(2))) float v2f;
typedef __attribute__((ext_vector_type(8))) float v8f;

#define BB    8
#define NN    2048
#define FIN   256
#define FOUT  64
#define NT    (NN / 16)
#define ALPHA_ 0.2f
#define NEG_BIG_ (-9000000000000000.0f)

// Low 32 bits of a generic pointer to LDS == wave-relative LDS byte offset
// (aperture lives in the high 32 bits; hardware uses only the low 32 bits as the LDS address).
static __device__ __forceinline__ unsigned lds_off(const void* p) {
  return (unsigned)(size_t)p;
}

// -------------------------------------------------------------------------
// Phase 1: Wh = h @ W  (16384x256 @ 256x64) via V_WMMA_F32_16X16X4_F32,
// plus e_src = Wh@a1, e_dst = Wh@a2 (butterfly reductions).
// W (64 KB) staged into LDS with async-to-LDS DMA.
// A layout (ISA 7.12.2, 32-bit A 16x4): lanes0-15 = M (dup 16-31);
//   V0 = K = k+2*lh, V1 = K = k+2*lh+1  (lh = lane>=16).
// B layout (4x16): V0 = row k+2*lh, V1 = row k+2*lh+1, column = lane%16.
// -------------------------------------------------------------------------
__global__ __launch_bounds__(128) void gat_phase1(
    const float* __restrict__ h, const float* __restrict__ W,
    float* __restrict__ wh, float* __restrict__ esrc, float* __restrict__ edst,
    const float* __restrict__ avec)
{
  __shared__ float Wl[FIN * FOUT];          // 64 KB
  const int t = threadIdx.x;
  {
    const unsigned l0 = lds_off(&Wl[t * 4]);        // thread's first float4
    const unsigned g0 = (unsigned)(t * 16);         // byte offset into W
    #pragma unroll
    for (int it = 0; it < 32; ++it) {
      asm volatile("global_load_async_to_lds_b128 %0, %1, %2"
                   :: "v"(l0 + (unsigned)(it * 2048)),
                      "v"(g0 + (unsigned)(it * 2048)), "s"(W) : "memory");
    }
    asm volatile("s_wait_asynccnt 0x0" ::: "memory");
  }
  __syncthreads();

  const int wave = t >> 5;
  const int ln   = t & 31;
  const int row  = ln & 15;
  const int lh   = ln >> 4;
  const int i0   = (blockIdx.x * 4 + wave) * 16;   // global row (b*N + n)

  v8f acc[4] = {};
  const float* hrow = h + (size_t)(i0 + row) * FIN;

  for (int k = 0; k < FIN; k += 4) {
    const v2f av = *(const v2f*)&hrow[k + 2 * lh];
    const int kr0 = (k + 2 * lh) * FOUT;
    #pragma unroll
    for (int ot = 0; ot < 4; ++ot) {
      v2f bv;
      bv.x = Wl[kr0 + ot * 16 + row];
      bv.y = Wl[kr0 + FOUT + ot * 16 + row];
      acc[ot] = __builtin_amdgcn_wmma_f32_16x16x4_f32(
          false, av, false, bv, (short)0, acc[ot], false, false);
    }
  }

  float a1o[4], a2o[4];
  #pragma unroll
  for (int ot = 0; ot < 4; ++ot) {
    a1o[ot] = avec[ot * 16 + row];
    a2o[ot] = avec[FOUT + ot * 16 + row];
  }

  #pragma unroll
  for (int v = 0; v < 8; ++v) {
    float es = 0.f, ed = 0.f;
    #pragma unroll
    for (int ot = 0; ot < 4; ++ot) {
      es += acc[ot][v] * a1o[ot];
      ed += acc[ot][v] * a2o[ot];
    }
    #pragma unroll
    for (int mk = 8; mk >= 1; mk >>= 1) {
      es += __shfl_xor(es, mk, 32);
      ed += __shfl_xor(ed, mk, 32);
    }
    const int grow = i0 + v + 8 * lh;
    if (row == 0) { esrc[grow] = es; edst[grow] = ed; }
    #pragma unroll
    for (int ot = 0; ot < 4; ++ot)
      wh[(size_t)grow * FOUT + ot * 16 + row] = acc[ot][v];
  }
}

// -------------------------------------------------------------------------
// Phase 2: flash-style masked softmax + P @ Wh.
// Double-buffered LDS; adj (128 MB, read-once -> th:TH_LOAD_NT) and the
// shared Wh j-tile staged by GLOBAL_LOAD_ASYNC_TO_LDS_B128 (ASYNCcnt),
// one barrier per j-tile, async loads overlap a full compute iteration.
// -------------------------------------------------------------------------
__global__ __launch_bounds__(128) void gat_phase2(
    const int* __restrict__ adj, const float* __restrict__ wh,
    const float* __restrict__ esrc, const float* __restrict__ edst,
    float* __restrict__ out)
{
  __shared__ int   adjt[2][64 * 16];        // 2 x 4 KB
  __shared__ float wht[2][16 * FOUT];       // 2 x 4 KB

  const int t    = threadIdx.x;
  const int wave = t >> 5;
  const int ln   = t & 31;
  const int row  = ln & 15;
  const int lh   = ln >> 4;
  const int b    = blockIdx.x >> 5;         // 32 blocks per batch
  const int br0  = (blockIdx.x & 31) * 64;  // block's first row in batch
  const int i0   = br0 + wave * 16;
  const size_t gr0 = (size_t)b * NN + i0;

  // cooperative staging mapping
  const int arow = t >> 1;                  // adj row 0..63
  const int acg  = (t & 1) * 8;             // adj col group (2 b128 per thread)
  const int wjj  = t >> 4;                  // wh row 0..7 (and +8 via offset:2048)
  const int wf4  = t & 15;                  // wh float4 within row
  const unsigned adj_lds[2] = { lds_off(&adjt[0][arow * 16 + acg]),
                                lds_off(&adjt[1][arow * 16 + acg]) };
  const unsigned wht_lds[2] = { lds_off(&wht[0][wjj * FOUT + wf4 * 4]),
                                lds_off(&wht[1][wjj * FOUT + wf4 * 4]) };
  const size_t adj_gbase = (((size_t)b * NN + br0 + arow) * NN + acg) * 4;
  const size_t wht_gbase = (((size_t)b * NN + wjj) * FOUT + wf4 * 4) * 4;

  auto stage = [&](int buf, int j0) {
    const unsigned ga = (unsigned)(adj_gbase + (size_t)j0 * 4);
    const unsigned gw = (unsigned)(wht_gbase + (size_t)j0 * FOUT * 4);
    asm volatile(
        "global_load_async_to_lds_b128 %0, %2, %4 th:TH_LOAD_NT\n\t"
        "global_load_async_to_lds_b128 %0, %2, %4 offset:16 th:TH_LOAD_NT\n\t"
        "global_load_async_to_lds_b128 %1, %3, %5\n\t"
        "global_load_async_to_lds_b128 %1, %3, %5 offset:2048"
        :: "v"(adj_lds[buf]), "v"(wht_lds[buf]), "v"(ga), "v"(gw),
           "s"(adj), "s"(wh)
        : "memory");
  };

  const float es = esrc[gr0 + row];
  float m = -3.0e38f, l = 0.f;
  v8f acc[4] = {};

  stage(0, 0);
  for (int jt = 0; jt < NT; ++jt) {
    asm volatile("s_wait_asynccnt 0x0" ::: "memory");  // tile jt landed
    __syncthreads();                                   // visible to all waves
    const int buf = jt & 1;
    if (jt + 1 < NT) stage(buf ^ 1, (jt + 1) * 16);    // overlap next tile
    const int j0 = jt * 16;

    // masked leaky-relu scores in WMMA A layout:
    // lane (row,lh) holds j = 4c + 2*lh + u  ->  p[2c+u]
    float p[8];
    float mt = -3.0e38f;
    #pragma unroll
    for (int c = 0; c < 4; ++c) {
      #pragma unroll
      for (int u = 0; u < 2; ++u) {
        const int jl = 4 * c + 2 * lh + u;
        float e = es + edst[(size_t)b * NN + j0 + jl];
        e = e > 0.f ? e : ALPHA_ * e;
        const float s = (adjt[buf][(wave * 16 + row) * 16 + jl] > 0) ? e : NEG_BIG_;
        p[2 * c + u] = s;
        mt = fmaxf(mt, s);
      }
    }
    // lanes r and r+16 jointly cover the 16 j's of row r
    mt = fmaxf(mt, __shfl_xor(mt, 16, 32));
    const float mnew  = fmaxf(m, mt);
    const float scale = __expf(m - mnew);
    float ls = 0.f;
    #pragma unroll
    for (int q = 0; q < 8; ++q) { p[q] = __expf(p[q] - mnew); ls += p[q]; }
    ls += __shfl_xor(ls, 16, 32);
    l = l * scale + ls;
    m = mnew;

    // rescale accumulators: acc row = v + 8*lh, scale lives in lane (v+8*lh)
    #pragma unroll
    for (int v = 0; v < 8; ++v) {
      const float sv = __shfl(scale, v + 8 * lh, 32);
      #pragma unroll
      for (int ot = 0; ot < 4; ++ot) acc[ot][v] *= sv;
    }

    // P(16x16) @ Wh_tile(16x64): 4 K-chunks x 4 o-tiles of f32 WMMA
    #pragma unroll
    for (int c = 0; c < 4; ++c) {
      v2f av; av.x = p[2 * c]; av.y = p[2 * c + 1];
      #pragma unroll
      for (int ot = 0; ot < 4; ++ot) {
        v2f bv;
        bv.x = wht[buf][(4 * c + 2 * lh) * FOUT + ot * 16 + row];
        bv.y = wht[buf][(4 * c + 2 * lh + 1) * FOUT + ot * 16 + row];
        acc[ot] = __builtin_amdgcn_wmma_f32_16x16x4_f32(
            false, av, false, bv, (short)0, acc[ot], false, false);
      }
    }
  }

  // epilogue: normalize, ELU, store
  #pragma unroll
  for (int v = 0; v < 8; ++v) {
    const float lv = __shfl(l, v + 8 * lh, 32);
    const float inv = 1.f / lv;
    #pragma unroll
    for (int ot = 0; ot < 4; ++ot) {
      float x = acc[ot][v] * inv;
      x = x > 0.f ? x : (__expf(x) - 1.f);
      out[(gr0 + v + 8 * lh) * FOUT + ot * 16 + row] = x;
    }
  }
}

extern "C" void kernel_launch(void* const* d_in, const int* in_sizes, int n_in,
                              void* d_out, int out_size, void* d_ws, size_t ws_size,
                              hipStream_t stream) {
  (void)in_sizes; (void)n_in; (void)out_size; (void)ws_size;
  const float* h   = (const float*)d_in[0];   // (8,2048,256) f32
  const int*   adj = (const int*)  d_in[1];   // (8,2048,2048) i32
  const float* W   = (const float*)d_in[2];   // (256,64) f32
  const float* a   = (const float*)d_in[3];   // (128,1) f32
  float* out = (float*)d_out;                 // (8,2048,64) f32

  float* wh   = (float*)d_ws;                          // 4 MB
  float* esrc = wh + (size_t)BB * NN * FOUT;           // 64 KB
  float* edst = esrc + (size_t)BB * NN;                // 64 KB

  gat_phase1<<<(BB * NN) / (4 * 16), 128, 0, stream>>>(h, W, wh, esrc, edst, a);
  gat_phase2<<<(BB * NN) / (4 * 16), 128, 0, stream>>>(adj, wh, esrc, edst, out);
}